// KDMSEMinimaxRelerrLoss_27616639714054
// MI455X (gfx1250) — compile-verified
//
#include <hip/hip_runtime.h>
#include <math.h>

typedef float v2f __attribute__((ext_vector_type(2)));
typedef float v8f __attribute__((ext_vector_type(8)));

#define VDIM   8192
#define ROWS   256                 // B*S = 2*128
#define NELEM  (ROWS * VDIM)       // 2097152
#define NGROUP (NELEM / 16)        // 131072 groups of 16 elements
#define NPV    10
#define NTILE  25                  // 25 tiles * 16 gammas = 400
#define MAIN_BLOCKS 1024

// ws layout (floats): [0,256) rowMax | [256,512) rowSumExp | [512, 512+blocks) partials
#define WS_ROWMAX 0
#define WS_ROWSUM 256
#define WS_PART   512

// ---------------- row-wise softmax statistics (max, sum of exp) ----------------
__global__ __launch_bounds__(256) void rowstat_kernel(const float* __restrict__ te,
                                                      float* __restrict__ ws) {
    __shared__ float sm[256];
    const int row = blockIdx.x;
    const float* p = te + (size_t)row * VDIM;
    float m = -__builtin_inff();
    for (int i = threadIdx.x; i < VDIM; i += 256) m = fmaxf(m, p[i]);
    sm[threadIdx.x] = m; __syncthreads();
    for (int s = 128; s > 0; s >>= 1) {
        if (threadIdx.x < s) sm[threadIdx.x] = fmaxf(sm[threadIdx.x], sm[threadIdx.x + s]);
        __syncthreads();
    }
    const float rmax = sm[0]; __syncthreads();
    float acc = 0.f;
    for (int i = threadIdx.x; i < VDIM; i += 256) acc += expf(p[i] - rmax);
    sm[threadIdx.x] = acc; __syncthreads();
    for (int s = 128; s > 0; s >>= 1) {
        if (threadIdx.x < s) sm[threadIdx.x] += sm[threadIdx.x + s];
        __syncthreads();
    }
    if (threadIdx.x == 0) { ws[WS_ROWMAX + row] = rmax; ws[WS_ROWSUM + row] = sm[0]; }
}

// ---------------- main: WMMA gamma-grid search + KD-MSE partial sums ----------------
__global__ __launch_bounds__(256) void kd_main_kernel(const float* __restrict__ st,
                                                      const float* __restrict__ te,
                                                      const int*   __restrict__ tgt,
                                                      const float* __restrict__ ws,
                                                      float*       __restrict__ partials) {
    const int tid   = threadIdx.x;
    const int lane  = tid & 31;
    const int n     = lane & 15;           // element-within-group AND gamma column
    const bool loHalf = (lane < 16);
    const int gwave = blockIdx.x * 8 + (tid >> 5);
    const int nwave = gridDim.x * 8;

    float accLoss = 0.f;

    for (int grp = gwave; grp < NGROUP; grp += nwave) {   // wave-uniform loop
        const int e    = (grp << 4) + n;                  // both halves duplicate elem n
        const int row  = e >> 13;
        const float t  = te[e];
        const float s  = st[e];
        const float rmax = ws[WS_ROWMAX + row];
        const float rsum = ws[WS_ROWSUM + row];
        const float phat = expf(t - rmax) / rsum;
        const float lph  = logf(phat);
        const float bl   = fmaxf(phat * (1.0f / 3.0f), 1e-6f);   // phat/(1+C), C=2
        const float bh   = fminf(phat * 3.0f, 1.0f);             // phat*(1+C)

        // Per-j quadratic coefficients; packed directly into WMMA A-operand layout:
        // lanes<16 carry (a_j, b_j) = K0,K1 ; lanes>=16 carry (c_j, 0) = K2,K3.
        float Alo[NPV], Ahi[NPV];
#pragma unroll
        for (int j = 0; j < NPV; ++j) {
            const float tj  = (float)j * (1.0f / 9.0f);
            const float pv  = bl + tj * (bh - bl);
            const float dp  = pv - phat;
            const float dl  = logf(pv) - lph;
            const float var = pv * (1.0f - pv);
            const float a = fmaf(dp, dp, var);            // dp^2 + var
            const float b = 2.0f * (var - dp * dl);       // -2 dp dl + 2 var
            const float c = fmaf(dl, dl, var);            // dl^2 + var
            Alo[j] = loHalf ? a : c;
            Ahi[j] = loHalf ? b : 0.0f;
        }

        float bestV[8], bestI[8];
#pragma unroll
        for (int r = 0; r < 8; ++r) { bestV[r] = __builtin_inff(); bestI[r] = 0.0f; }

        for (int T = 0; T < NTILE; ++T) {
            const int   gi = T * 16 + n;
            const float g  = -10.0f + 0.05f * (float)gi;
            // B-operand: K0,K1 rows (g^2, g) in lanes<16 ; K2,K3 rows (1, 0) in lanes>=16
            const v2f B = { loHalf ? g * g : 1.0f, loHalf ? g : 0.0f };

            v8f mx;
#pragma unroll
            for (int j = 0; j < NPV; ++j) {
                const v2f A = { Alo[j], Ahi[j] };
                v8f zero = {0.f, 0.f, 0.f, 0.f, 0.f, 0.f, 0.f, 0.f};
                // D[m][n] = a_j[m]*g_n^2 + b_j[m]*g_n + c_j[m]
                v8f d = __builtin_amdgcn_wmma_f32_16x16x4_f32(
                            false, A, false, B, (short)0, zero, false, false);
                if (j == 0) mx = d;
                else {
#pragma unroll
                    for (int r = 0; r < 8; ++r) mx[r] = fmaxf(mx[r], d[r]);
                }
            }
            const float gif = (float)gi;
#pragma unroll
            for (int r = 0; r < 8; ++r) {          // strict '<' keeps earliest gamma
                const bool lt = mx[r] < bestV[r];
                bestV[r] = lt ? mx[r] : bestV[r];
                bestI[r] = lt ? gif   : bestI[r];
            }
        }

        // argmin across the 16 gamma-columns of each half (butterfly, low-index ties)
#pragma unroll
        for (int r = 0; r < 8; ++r) {
#pragma unroll
            for (int x = 1; x <= 8; x <<= 1) {
                const float ov = __shfl_xor(bestV[r], x, 32);
                const float oi = __shfl_xor(bestI[r], x, 32);
                const bool take = (ov < bestV[r]) || (ov == bestV[r] && oi < bestI[r]);
                bestV[r] = take ? ov : bestV[r];
                bestI[r] = take ? oi : bestI[r];
            }
        }
        // row r now holds: elem r (lanes<16) / elem r+8 (lanes>=16). Route elem (lane&15)
        // back to each lane: half-swap + 3-level mux on (lane&7).
        float sel[8];
#pragma unroll
        for (int r = 0; r < 8; ++r) {
            const float sw = __shfl_xor(bestI[r], 16, 32);
            const bool own = (((lane >> 3) & 1) == ((lane >> 4) & 1));
            sel[r] = own ? bestI[r] : sw;
        }
        const int rsel = lane & 7;
        const float m0 = (rsel & 1) ? sel[1] : sel[0];
        const float m1 = (rsel & 1) ? sel[3] : sel[2];
        const float m2 = (rsel & 1) ? sel[5] : sel[4];
        const float m3 = (rsel & 1) ? sel[7] : sel[6];
        const float q0 = (rsel & 2) ? m1 : m0;
        const float q1 = (rsel & 2) ? m3 : m2;
        const float idxF = (rsel & 4) ? q1 : q0;

        const float gamma = -10.0f + 0.05f * idxF;
        const float y     = ((e & (VDIM - 1)) == tgt[row]) ? 1.0f : 0.0f;
        const float resid = s - (t + gamma * (y - phat));
        accLoss = fmaf(resid, resid, accLoss);      // both halves add -> 2x true sum
    }

    __shared__ float sm[256];
    sm[tid] = accLoss; __syncthreads();
    for (int s2 = 128; s2 > 0; s2 >>= 1) {
        if (tid < s2) sm[tid] += sm[tid + s2];
        __syncthreads();
    }
    if (tid == 0) partials[blockIdx.x] = sm[0];
}

// ---------------- deterministic final reduction ----------------
__global__ void finalize_kernel(const float* __restrict__ partials, int nparts,
                                const float* __restrict__ loss_orig,
                                float* __restrict__ out) {
    if (threadIdx.x == 0 && blockIdx.x == 0) {
        float s = 0.f;
        for (int i = 0; i < nparts; ++i) s += partials[i];
        const float loss_kd = s / (2.0f * (float)NELEM);  // /2: halves double-count
        out[0] = 0.5f * loss_orig[0] + 0.5f * loss_kd;    // ALPHA = 0.5
    }
}

extern "C" void kernel_launch(void* const* d_in, const int* in_sizes, int n_in,
                              void* d_out, int out_size, void* d_ws, size_t ws_size,
                              hipStream_t stream) {
    const float* st  = (const float*)d_in[0];
    const float* te  = (const float*)d_in[1];
    const int*   tgt = (const int*)d_in[2];
    const float* lo  = (const float*)d_in[3];
    float* ws = (float*)d_ws;

    int blocks = MAIN_BLOCKS;
    const size_t needF = (size_t)WS_PART + (size_t)blocks;
    if (ws_size < needF * sizeof(float)) {
        long avail = (long)(ws_size / sizeof(float)) - WS_PART;
        blocks = (avail < 1) ? 1 : (int)avail;
        if (blocks > MAIN_BLOCKS) blocks = MAIN_BLOCKS;
    }

    rowstat_kernel<<<ROWS, 256, 0, stream>>>(te, ws);
    kd_main_kernel<<<blocks, 256, 0, stream>>>(st, te, tgt, ws, ws + WS_PART);
    finalize_kernel<<<1, 32, 0, stream>>>(ws + WS_PART, blocks, lo, (float*)d_out);
}